// PathSampler_5789615915254
// MI455X (gfx1250) — compile-verified
//
#include <hip/hip_runtime.h>

// ---------------------------------------------------------------------------
// Path sampler for MI455X (gfx1250, wave32).
// One wave = 32 rays. MLP done with V_WMMA_F32_16X16X4_F32 in hidden-major
// orientation so all operand marshaling is static half-wave lane exchanges
// (ds_bpermute), no LDS, no barriers. Grid (8MB) is L2-resident; kernel is
// latency bound on the 192-step serial chain -> 1024 waves hide it.
// ReLU emitted as single v_med3_num_f32 (x clamped to [0,+inf]).
// ---------------------------------------------------------------------------

typedef float v2f __attribute__((ext_vector_type(2)));
typedef float v8f __attribute__((ext_vector_type(8)));

#define NSIDE   128
#define STEPF   (2.0f / 192.0f)
#define GSCALE  63.5f              // (128-1)/(1-(-1))

__device__ __forceinline__ float bperm(float v, unsigned srcLane) {
    // dst[lane] = v[srcLane]; EXEC is all-ones at every call site.
    return __int_as_float(
        __builtin_amdgcn_ds_bpermute((int)(srcLane << 2), __float_as_int(v)));
}

__device__ __forceinline__ float relu1(float x) {
    // v_med3_num_f32 x, 0, +inf  ==  clamp to [0, inf)  ==  ReLU, 1 VALU op
    return __builtin_amdgcn_fmed3f(x, 0.0f, __builtin_inff());
}

__device__ __forceinline__ v8f wmma4(v2f a, v2f b, v8f c) {
    // D(16x16,f32) = A(16x4,f32) * B(4x16,f32) + C
    return __builtin_amdgcn_wmma_f32_16x16x4_f32(
        /*neg_a=*/false, a, /*neg_b=*/false, b,
        /*c_mod=*/(short)0, c, /*reuse_a=*/false, /*reuse_b=*/false);
}

__global__ __launch_bounds__(256) void path_sampler_wmma(
    const float* __restrict__ origin, const float* __restrict__ dir_in,
    const float* __restrict__ grid,   const float* __restrict__ W1,
    const float* __restrict__ b1,     const float* __restrict__ W2,
    const float* __restrict__ b2,     const float* __restrict__ alpha_p,
    const int*   __restrict__ ns_p,   float* __restrict__ out, int B)
{
    const unsigned lane  = threadIdx.x & 31u;
    const unsigned wave  = threadIdx.x >> 5;
    const int      ray   = (int)((blockIdx.x * 8u + wave) * 32u + lane);
    if (ray >= B) return;                       // B is a multiple of 256

    const bool     lo    = lane < 16u;
    const unsigned l15   = lane & 15u;
    const unsigned lxor  = lane ^ 16u;
    const float    alpha = alpha_p[0];
    const int      ns    = ns_p[0];

    // d_out layout: pos[B,ns,3] | dir[B,ns,3] | dist[B,ns] | data[B,ns] | grad[B,ns,3]
    float* out_pos  = out;
    float* out_dir  = out + (size_t)B * ns * 3;
    float* out_dist = out + (size_t)B * ns * 6;
    float* out_data = out_dist + (size_t)B * ns;
    float* out_grad = out_data + (size_t)B * ns;

    // ---- loop-invariant WMMA operands --------------------------------------
    // Layer 1: A = W1^T tiles (hidden rows, K=coord padded to 4)
    v2f a1[4];  v8f cb[4];
    #pragma unroll
    for (int t = 0; t < 4; ++t) {
        const int hid = 16 * t + (int)l15;
        a1[t].x = lo ? W1[hid]        : W1[128 + hid];   // K=0 | K=2
        a1[t].y = lo ? W1[64 + hid]   : 0.0f;            // K=1 | K=3 (pad)
        #pragma unroll
        for (int j = 0; j < 8; ++j)
            cb[t][j] = lo ? b1[16 * t + j] : b1[16 * t + 8 + j];
    }
    // Layer 2: A = W2^T chunks (M=pred comp 0..2 of 16, K=4 hidden per chunk)
    v2f a2[16];
    #pragma unroll
    for (int k = 0; k < 16; ++k) {
        const int m  = (int)l15;
        const int k0 = 4 * k + (lo ? 0 : 2);
        a2[k].x = (m < 3) ? W2[(k0 + 0) * 3 + m] : 0.0f;
        a2[k].y = (m < 3) ? W2[(k0 + 1) * 3 + m] : 0.0f;
    }
    v8f d2i;
    #pragma unroll
    for (int j = 0; j < 8; ++j) d2i[j] = (j < 3 && lo) ? b2[j] : 0.0f;

    // ---- ray state ---------------------------------------------------------
    float px = origin[3 * ray + 0], py = origin[3 * ray + 1], pz = origin[3 * ray + 2];
    float dx0 = dir_in[3 * ray + 0], dy0 = dir_in[3 * ray + 1], dz0 = dir_in[3 * ray + 2];
    float rdx = dx0, rdy = dy0, rdz = dz0;
    float rt = 0.0f;

    const size_t rb = (size_t)ray * ns;
    out_pos[rb * 3 + 0] = px; out_pos[rb * 3 + 1] = py; out_pos[rb * 3 + 2] = pz;
    {
        float n = rsqrtf(fmaxf(dx0 * dx0 + dy0 * dy0 + dz0 * dz0, 1e-10f));
        out_dir[rb * 3 + 0] = dx0 * n; out_dir[rb * 3 + 1] = dy0 * n; out_dir[rb * 3 + 2] = dz0 * n;
    }
    out_dist[rb] = 0.0f;

    // ---- main serial march -------------------------------------------------
    for (int s = 0; s < ns; ++s) {
        // trilinear value + gradient (grid is L2-resident)
        float gx = fminf(fmaxf((px + 1.0f) * GSCALE, 0.0f), 127.0f);
        float gy = fminf(fmaxf((py + 1.0f) * GSCALE, 0.0f), 127.0f);
        float gz = fminf(fmaxf((pz + 1.0f) * GSCALE, 0.0f), 127.0f);
        int ix = (int)gx; ix = ix > 126 ? 126 : ix;
        int iy = (int)gy; iy = iy > 126 ? 126 : iy;
        int iz = (int)gz; iz = iz > 126 ? 126 : iz;
        float fx = gx - (float)ix, fy = gy - (float)iy, fz = gz - (float)iz;
        const float* gp = grid + ((ix * 128 + iy) * 128 + iz);
        float c000 = gp[0],     c001 = gp[1],     c010 = gp[128],   c011 = gp[129];
        float c100 = gp[16384], c101 = gp[16385], c110 = gp[16512], c111 = gp[16513];
        float c00 = c000 * (1.f - fx) + c100 * fx, c01 = c001 * (1.f - fx) + c101 * fx;
        float c10 = c010 * (1.f - fx) + c110 * fx, c11 = c011 * (1.f - fx) + c111 * fx;
        float c0 = c00 * (1.f - fy) + c10 * fy,    c1 = c01 * (1.f - fy) + c11 * fy;
        float val = c0 * (1.f - fz) + c1 * fz;
        float gdx = (((c100-c000)*(1.f-fy) + (c110-c010)*fy)*(1.f-fz)
                   + ((c101-c001)*(1.f-fy) + (c111-c011)*fy)*fz) * GSCALE;
        float gdy = (((c010-c000)*(1.f-fx) + (c110-c100)*fx)*(1.f-fz)
                   + ((c011-c001)*(1.f-fx) + (c111-c101)*fx)*fz) * GSCALE;
        float gdz = (((c001-c000)*(1.f-fx) + (c011-c010)*fx)*(1.f-fy)
                   + ((c101-c100)*(1.f-fx) + (c111-c110)*fx)*fy) * GSCALE;

        // ---- layer 1: H^T(64x32) = W1^T(64x3) * RP^T(3x32) via 8 WMMAs ----
        float xs0 = bperm(px, l15),      ys0 = bperm(py, l15),      zs0 = bperm(pz, l15);
        float xs1 = bperm(px, l15 + 16), ys1 = bperm(py, l15 + 16), zs1 = bperm(pz, l15 + 16);
        v2f bt0, bt1;
        bt0.x = lo ? xs0 : zs0;  bt0.y = lo ? ys0 : 0.0f;   // rays 0..15
        bt1.x = lo ? xs1 : zs1;  bt1.y = lo ? ys1 : 0.0f;   // rays 16..31
        v8f h[4][2];
        #pragma unroll
        for (int t = 0; t < 4; ++t) {
            h[t][0] = wmma4(a1[t], bt0, cb[t]);
            h[t][1] = wmma4(a1[t], bt1, cb[t]);
        }
        #pragma unroll
        for (int t = 0; t < 4; ++t)
            #pragma unroll
            for (int c = 0; c < 2; ++c)
                #pragma unroll
                for (int j = 0; j < 8; ++j)
                    h[t][c][j] = relu1(h[t][c][j]);   // 1x v_med3_num each

        // ---- layer 2: pred^T(3x32) = W2^T(3x64) * H via 32 WMMAs ----------
        v8f d0 = d2i, d1 = d2i;
        #pragma unroll
        for (int k = 0; k < 16; ++k) {
            const int  t    = k >> 2;
            const int  base = (k & 1) ? 4 : 0;
            const bool swl  = (k & 2) != 0;
            {   // rays 0..15
                v8f ht = h[t][0];
                float v0 = ht[base], v1 = ht[base+1], v2 = ht[base+2], v3 = ht[base+3];
                float xa = bperm(swl ? v0 : v2, lxor);
                float xb = bperm(swl ? v1 : v3, lxor);
                v2f bb;
                bb.x = swl ? (lo ? xa : v2) : (lo ? v0 : xa);
                bb.y = swl ? (lo ? xb : v3) : (lo ? v1 : xb);
                d0 = wmma4(a2[k], bb, d0);
            }
            {   // rays 16..31
                v8f ht = h[t][1];
                float v0 = ht[base], v1 = ht[base+1], v2 = ht[base+2], v3 = ht[base+3];
                float xa = bperm(swl ? v0 : v2, lxor);
                float xb = bperm(swl ? v1 : v3, lxor);
                v2f bb;
                bb.x = swl ? (lo ? xa : v2) : (lo ? v0 : xa);
                bb.y = swl ? (lo ? xb : v3) : (lo ? v1 : xb);
                d1 = wmma4(a2[k], bb, d1);
            }
        }
        // extract this lane's pred components (comp c lives in VGPR c, lanes 0..15)
        float p0a = bperm(d0[0], l15), p0b = bperm(d1[0], l15);
        float p1a = bperm(d0[1], l15), p1b = bperm(d1[1], l15);
        float p2a = bperm(d0[2], l15), p2b = bperm(d1[2], l15);
        float prx = (lo ? p0a : p0b) * alpha;
        float pry = (lo ? p1a : p1b) * alpha;
        float prz = (lo ? p2a : p2b) * alpha;

        // grad = where(||idx_grad|| > 1e-3, pred_grad, idx_grad)
        float g2  = gdx * gdx + gdy * gdy + gdz * gdz;
        bool  big = g2 > 1e-6f;
        float sgx = big ? prx : gdx, sgy = big ? pry : gdy, sgz = big ? prz : gdz;

        // state update
        float inv = STEPF / val;
        float dpx = inv * rdx, dpy = inv * rdy, dpz = inv * rdz;
        float npx = px + dpx, npy = py + dpy, npz = pz + dpz;
        float nrx = rdx + STEPF * sgx, nry = rdy + STEPF * sgy, nrz = rdz + STEPF * sgz;
        float nrt = rt + sqrtf(dpx * dpx + dpy * dpy + dpz * dpz);

        // outputs for this step
        out_data[rb + s] = val;
        out_grad[(rb + s) * 3 + 0] = gdx;
        out_grad[(rb + s) * 3 + 1] = gdy;
        out_grad[(rb + s) * 3 + 2] = gdz;
        if (s + 1 < ns) {                     // wave-uniform branch
            const size_t o = rb + s + 1;
            out_pos[o * 3 + 0] = npx; out_pos[o * 3 + 1] = npy; out_pos[o * 3 + 2] = npz;
            float n = rsqrtf(fmaxf(nrx * nrx + nry * nry + nrz * nrz, 1e-10f));
            out_dir[o * 3 + 0] = nrx * n; out_dir[o * 3 + 1] = nry * n; out_dir[o * 3 + 2] = nrz * n;
            out_dist[o] = nrt;
        }
        px = npx; py = npy; pz = npz;
        rdx = nrx; rdy = nry; rdz = nrz;
        rt = nrt;
    }
}

extern "C" void kernel_launch(void* const* d_in, const int* in_sizes, int n_in,
                              void* d_out, int out_size, void* d_ws, size_t ws_size,
                              hipStream_t stream) {
    const float* origin    = (const float*)d_in[0];
    const float* direction = (const float*)d_in[1];
    const float* grid      = (const float*)d_in[2];
    const float* W1        = (const float*)d_in[3];
    const float* b1        = (const float*)d_in[4];
    const float* W2        = (const float*)d_in[5];
    const float* b2        = (const float*)d_in[6];
    const float* alpha     = (const float*)d_in[7];
    const int*   nsamp     = (const int*)d_in[8];
    const int B = in_sizes[0] / 3;                 // 32768
    dim3 block(256);
    dim3 gridd((B + 255) / 256);                   // 128 blocks = 1024 waves
    path_sampler_wmma<<<gridd, block, 0, stream>>>(
        origin, direction, grid, W1, b1, W2, b2, alpha, nsamp,
        (float*)d_out, B);
}